// AccumAtt_81398220194306
// MI455X (gfx1250) — compile-verified
//
#include <hip/hip_runtime.h>
#include <hip/hip_bf16.h>
#include <math.h>

// ---------------------------------------------------------------------------
// AccumAtt (TAM-style temporal attention) for MI455X / gfx1250, wave32.
//
// Shapes: x [NT=128, C=512, HW=784] f32.  c8 = 64, B = 16, T = 8.
// Bandwidth-bound: ~616 MB HBM traffic -> ~26 us floor @ 23.3 TB/s.
// Structure: mean pass -> WMMA f32 GEMM (+BN+ReLU) -> tiny scan -> scale pass.
// ---------------------------------------------------------------------------

#define NT   128
#define CCH  512
#define HW   784      // 196 float4
#define HW4  196
#define C8   64
#define BB   16
#define TT   8
#define EPSV 1e-5f

typedef float v2f __attribute__((ext_vector_type(2)));
typedef float v8f __attribute__((ext_vector_type(8)));

static __device__ __forceinline__ float wave_sum32(float v) {
    #pragma unroll
    for (int off = 16; off > 0; off >>= 1)
        v += __shfl_xor(v, off, 32);
    return v;
}

static __device__ __forceinline__ float sigm(float x) {
    return 1.0f / (1.0f + __expf(-x));
}

// ---------------------------------------------------------------------------
// Kernel 1: spatial mean.  One wave per (nt, c) row of 784 floats.
// Grid: 8192 blocks x 256 threads (8 waves/block, 8 rows/block).
// ---------------------------------------------------------------------------
__global__ __launch_bounds__(256)
void k_mean(const float* __restrict__ x, float* __restrict__ xv) {
    const int lane = threadIdx.x & 31;
    const int wv   = threadIdx.x >> 5;
    const int row  = blockIdx.x * 8 + wv;          // row = nt*512 + c
    const float4* xr = (const float4*)(x) + (size_t)row * HW4;

    float s = 0.0f;
    #pragma unroll
    for (int i = 0; i < 7; ++i) {
        int idx = lane + 32 * i;
        if (idx < HW4) {
            float4 v = xr[idx];
            s += (v.x + v.y) + (v.z + v.w);
        }
    }
    s = wave_sum32(s);
    if (lane == 0) xv[row] = s * (1.0f / (float)HW);
}

// ---------------------------------------------------------------------------
// Kernel 2: Y[128,128] = XV[128,512] @ [w1^T | w2^T], then BN + ReLU.
// V_WMMA_F32_16X16X4_F32, one wave per 16x16 tile, K stepped by 4.
//
// A 16x4 f32 layout (ISA 7.12.2): lanes 0-15: K=0 (v0), K=1 (v1);
//                                 lanes 16-31: K=2 (v0), K=3 (v1).
// B 4x16 f32 (row striped across lanes): lanes 0-15: rows K=0/K=1;
//                                        lanes 16-31: rows K=2/K=3.
// Per lane both A and B operands are 2 consecutive K values -> float2 loads
// (w1/w2 are [64,512] row-major so w[col][k],w[col][k+1] are contiguous).
// C/D 16x16 f32: vgpr r, lanes 0-15 -> M=r, N=lane; lanes 16-31 -> M=8+r.
// ---------------------------------------------------------------------------
__global__ __launch_bounds__(256)
void k_gemm(const float* __restrict__ xv,
            const float* __restrict__ w1, const float* __restrict__ w2,
            const float* __restrict__ bn1_g, const float* __restrict__ bn1_b,
            const float* __restrict__ bn1_m, const float* __restrict__ bn1_v,
            const float* __restrict__ bn2_g, const float* __restrict__ bn2_b,
            const float* __restrict__ bn2_m, const float* __restrict__ bn2_v,
            float* __restrict__ LR) {
    const int lane = threadIdx.x & 31;
    const int wv   = threadIdx.x >> 5;
    const int tile = blockIdx.x * 8 + wv;          // 0..63
    const int m0   = (tile >> 3) * 16;             // output row base (nt)
    const int n0   = (tile & 7) * 16;              // output col base (channel)
    const int hf   = lane >> 4;                    // 0: K pair {0,1}; 1: {2,3}
    const int lr   = lane & 15;

    const int   arow = m0 + lr;                    // A row this lane carries
    const int   col  = n0 + lr;                    // B/D column this lane carries
    const float* wrow = (col < C8) ? (w1 + (size_t)col * CCH)
                                   : (w2 + (size_t)(col - C8) * CCH);
    const float* arowp = xv + (size_t)arow * CCH + 2 * hf;
    const float* browp = wrow + 2 * hf;

    v8f acc = {0.f, 0.f, 0.f, 0.f, 0.f, 0.f, 0.f, 0.f};
    for (int k = 0; k < CCH; k += 4) {
        v2f a = *(const v2f*)(arowp + k);
        v2f b = *(const v2f*)(browp + k);
        acc = __builtin_amdgcn_wmma_f32_16x16x4_f32(
            /*neg_a=*/false, a, /*neg_b=*/false, b,
            /*c_mod=*/(short)0, acc, /*reuse_a=*/false, /*reuse_b=*/false);
    }

    // BN(channel=col) + ReLU epilogue (divergence OK: WMMA region is done).
    float g, bb, mm, vv;
    if (col < C8) { g = bn1_g[col]; bb = bn1_b[col]; mm = bn1_m[col]; vv = bn1_v[col]; }
    else { int j = col - C8; g = bn2_g[j]; bb = bn2_b[j]; mm = bn2_m[j]; vv = bn2_v[j]; }
    const float scale = g * rsqrtf(vv + EPSV);
    const float shift = bb - mm * scale;

    #pragma unroll
    for (int r = 0; r < 8; ++r) {
        const int mrow = m0 + hf * 8 + r;
        float y = fmaxf(acc[r] * scale + shift, 0.0f);
        LR[(size_t)mrow * 128 + col] = y;
    }
}

// ---------------------------------------------------------------------------
// Kernel 3: temporal-difference gated scan -> sig [B*T, 512].
// Single block of 512 threads (16 waves). Wave b computes batch b's gate.
// diff[b][t][j] = t<T-1 ? LR[b*T+t][j] - LR[b*T+t+1][64+j] : 1
// ---------------------------------------------------------------------------
__global__ __launch_bounds__(512)
void k_scan(const float* __restrict__ LR,
            const float* __restrict__ gamma_w, const float* __restrict__ gamma_b,
            const float* __restrict__ Wa_w, const float* __restrict__ Wa_b,
            float* __restrict__ sig) {
    __shared__ float gstate[BB * C8];
    __shared__ float snew[BB * C8];
    __shared__ float gate[BB];

    const int tid  = threadIdx.x;
    const int lane = tid & 31;
    const int wv   = tid >> 5;                      // batch for gate phase

    for (int idx = tid; idx < BB * C8; idx += 512) gstate[idx] = 1.0f;  // diff[:, -1]
    __syncthreads();

    for (int t = 0; t < TT; ++t) {
        // --- gate[b] = sigmoid( dot(cat(diff_t[b], gstate[b]), gamma_w) + gamma_b )
        float s = 0.0f;
        #pragma unroll
        for (int jj = lane; jj < 2 * C8; jj += 32) {
            float cv;
            if (jj < C8) {
                cv = (t < TT - 1)
                   ? (LR[(size_t)(wv * TT + t) * 128 + jj] -
                      LR[(size_t)(wv * TT + t + 1) * 128 + C8 + jj])
                   : 1.0f;
            } else {
                cv = gstate[wv * C8 + (jj - C8)];
            }
            s += cv * gamma_w[jj];
        }
        s = wave_sum32(s);
        if (lane == 0) gate[wv] = sigm(s + gamma_b[0]);
        __syncthreads();

        // --- new = diff*g + gstate*(1-g)
        for (int idx = tid; idx < BB * C8; idx += 512) {
            const int b = idx >> 6, j = idx & 63;
            const float g = gate[b];
            const float d = (t < TT - 1)
                ? (LR[(size_t)(b * TT + t) * 128 + j] -
                   LR[(size_t)(b * TT + t + 1) * 128 + C8 + j])
                : 1.0f;
            snew[idx] = d * g + gstate[idx] * (1.0f - g);
        }
        __syncthreads();

        // --- att[b][c] = sigmoid( new[b] . Wa_w[c] + Wa_b[c] ); thread c, loop b.
        const int c = tid;                          // 0..511
        const float* war = Wa_w + (size_t)c * C8;
        for (int b = 0; b < BB; ++b) {
            float s2 = Wa_b[c];
            #pragma unroll 8
            for (int j = 0; j < C8; ++j) s2 += snew[b * C8 + j] * war[j];
            sig[(size_t)(b * TT + t) * CCH + c] = sigm(s2);
        }
        __syncthreads();

        for (int idx = tid; idx < BB * C8; idx += 512) gstate[idx] = snew[idx];
        __syncthreads();
    }
}

// ---------------------------------------------------------------------------
// Kernel 4: out = x * sig (broadcast over HW). One wave per (nt, c) row.
// sig is stored [nt,c] flat == row index, so s = sig[row].
// ---------------------------------------------------------------------------
__global__ __launch_bounds__(256)
void k_scale(const float* __restrict__ x, const float* __restrict__ sig,
             float* __restrict__ out) {
    const int lane = threadIdx.x & 31;
    const int wv   = threadIdx.x >> 5;
    const int row  = blockIdx.x * 8 + wv;
    const float s  = sig[row];
    const float4* xr = (const float4*)(x) + (size_t)row * HW4;
    float4*       orr = (float4*)(out) + (size_t)row * HW4;

    #pragma unroll
    for (int i = 0; i < 7; ++i) {
        int idx = lane + 32 * i;
        if (idx < HW4) {
            float4 v = xr[idx];
            v.x *= s; v.y *= s; v.z *= s; v.w *= s;
            orr[idx] = v;
        }
    }
}

// ---------------------------------------------------------------------------
extern "C" void kernel_launch(void* const* d_in, const int* in_sizes, int n_in,
                              void* d_out, int out_size, void* d_ws, size_t ws_size,
                              hipStream_t stream) {
    const float* x      = (const float*)d_in[0];
    const float* w1     = (const float*)d_in[1];
    const float* bn1_g  = (const float*)d_in[2];
    const float* bn1_b  = (const float*)d_in[3];
    const float* bn1_m  = (const float*)d_in[4];
    const float* bn1_v  = (const float*)d_in[5];
    const float* w2     = (const float*)d_in[6];
    const float* bn2_g  = (const float*)d_in[7];
    const float* bn2_b  = (const float*)d_in[8];
    const float* bn2_m  = (const float*)d_in[9];
    const float* bn2_v  = (const float*)d_in[10];
    const float* Wa_w   = (const float*)d_in[11];
    const float* Wa_b   = (const float*)d_in[12];
    const float* gamma_w= (const float*)d_in[13];
    const float* gamma_b= (const float*)d_in[14];
    (void)d_in; (void)in_sizes; (void)n_in; (void)out_size; (void)ws_size;

    float* out = (float*)d_out;
    float* xv  = (float*)d_ws;                 // [128*512]
    float* LR  = xv + NT * CCH;                // [128*128]
    float* sig = LR + NT * 128;                // [128*512]

    const int rows = NT * CCH;                 // 65536 (nt, c) rows

    k_mean <<<rows / 8, 256, 0, stream>>>(x, xv);
    k_gemm <<<8,        256, 0, stream>>>(xv, w1, w2,
                                          bn1_g, bn1_b, bn1_m, bn1_v,
                                          bn2_g, bn2_b, bn2_m, bn2_v, LR);
    k_scan <<<1,        512, 0, stream>>>(LR, gamma_w, gamma_b, Wa_w, Wa_b, sig);
    k_scale<<<rows / 8, 256, 0, stream>>>(x, sig, out);
}